// StratifiedMinPooling_66314295050400
// MI455X (gfx1250) — compile-verified
//
#include <hip/hip_runtime.h>
#include <stdint.h>

// Stratified (label-grouped) min pooling for MI455X (gfx1250).
//
// out[b, c] = min_{j : labels[j] == c} values[b, j]
// B=16384, P=4096, C=512, fp32.
//
// Roofline: 256 MB values in + 32 MB out @ 23.3 TB/s  =>  ~12.4 us floor.
// Compute (67M min ops) is trivial => memory bound. WMMA cannot express
// `min`, so the CDNA5 paths used are:
//   * Tensor Data Mover: tensor_load_to_lds streams HBM->LDS 32KB tiles,
//     double-buffered, synchronized with s_wait_tensorcnt (TENSORcnt).
//   * ds_min_num_f32 LDS float-min atomics into a privatized per-block
//     accumulator tile (scatter reduction, 1 DS atomic per element).
//   * Non-temporal b128 output stores (write-once data, spare the 192MB L2).

#define B_DIM   16384
#define P_DIM   4096
#define C_DIM   512
#define ROWS    16            // rows of `values` per block
#define CHUNK   512           // columns per TDM tile
#define NCHUNK  (P_DIM / CHUNK)
#define THREADS 512           // 16 wave32s

typedef __attribute__((ext_vector_type(4))) unsigned int v4u;
typedef __attribute__((ext_vector_type(8))) int          v8i;
typedef __attribute__((ext_vector_type(4))) int          v4i;
typedef __attribute__((ext_vector_type(4))) float        v4f;

#if __has_include(<hip/amd_detail/amd_gfx1250_TDM.h>)
#define TDM_SIX_ARG 1
#else
#define TDM_SIX_ARG 0
#endif

// Issue one TDM load of a ROWS x CHUNK fp32 tile (row stride = P_DIM elems)
// from global memory into LDS at byte offset lds_off. Wave-level op, tracked
// with TENSORcnt.
__device__ __forceinline__ void tdm_load_tile(uint32_t lds_off, const float* gptr) {
  uint64_t ga = (uint64_t)(uintptr_t)gptr;

  // D# group 0 (128b): count=1 (valid), lds_addr, global_addr[56:0], type=2.
  v4u g0;
  g0.x = 1u;
  g0.y = lds_off;
  g0.z = (uint32_t)(ga & 0xFFFFFFFFu);
  g0.w = (uint32_t)((ga >> 32) & 0x01FFFFFFu) | (2u << 30);

  // D# group 1 (256b):
  //  w0: data_size=2 (4B), no multicast/iterate/pad/atomic-barrier.
  //  tensor_dim0 = P_DIM, tensor_dim1 = ROWS (tile fits exactly -> no OOB
  //  zero-fill, which would corrupt the min).
  //  tile_dim0 = CHUNK, tile_dim1 = ROWS, tile_dim2 = 0 (2-D tile).
  //  tensor_dim0_stride = P_DIM elements; dim1 stride unused.
  v8i g1;
  g1[0] = 0x00020000;                                         // data_size=4B
  g1[1] = (int)(((uint32_t)P_DIM & 0xFFFFu) << 16);           // dim0 lo16
  g1[2] = (int)(((uint32_t)P_DIM >> 16) |
                (((uint32_t)ROWS & 0xFFFFu) << 16));          // dim0 hi16 | dim1 lo16
  g1[3] = (int)(((uint32_t)CHUNK) << 16);                     // dim1 hi16(0) | tile_dim0
  g1[4] = (int)((uint32_t)ROWS);                              // tile_dim1 | tile_dim2=0
  g1[5] = (int)((uint32_t)P_DIM);                             // dim0_stride lo32
  g1[6] = 0;                                                  // dim0_stride hi16 | dim1_stride lo16
  g1[7] = 0;                                                  // dim1_stride hi32

  v4i g2 = {0, 0, 0, 0};   // dims 2/3 unused (tile_dim2 == 0)
  v4i g3 = {0, 0, 0, 0};

#if TDM_SIX_ARG
  v8i g4 = {0, 0, 0, 0, 0, 0, 0, 0};
  __builtin_amdgcn_tensor_load_to_lds(g0, g1, g2, g3, g4, 0);
#else
  __builtin_amdgcn_tensor_load_to_lds(g0, g1, g2, g3, 0);
#endif
}

__global__ __launch_bounds__(THREADS)
void stratified_min_kernel(const float* __restrict__ values,
                           const int* __restrict__ labels,
                           float* __restrict__ out) {
  // Privatized accumulator tile acc[r][c], +inf identity (segment_min).
  __shared__ float acc[ROWS * C_DIM];            // 32 KB
  // Double-buffered TDM staging for values tiles.
  __shared__ float stage[2][ROWS * CHUNK];       // 2 x 32 KB
  // 96 KB total -> 3 workgroups (48 wave32s) per 320KB WGP.

  const int tid  = threadIdx.x;
  const int row0 = blockIdx.x * ROWS;

  for (int i = tid; i < ROWS * C_DIM; i += THREADS)
    acc[i] = __builtin_inff();

  // LDS byte offsets (shared aperture lives in addr[63:32]; low 32 bits are
  // the workgroup-relative LDS offset).
  const uint32_t stage_off0 = (uint32_t)(uintptr_t)&stage[0][0];
  const uint32_t stage_off1 = (uint32_t)(uintptr_t)&stage[1][0];

  const bool issuer = (tid < 32);  // wave 0 drives the TDM

  if (issuer) {
    tdm_load_tile(stage_off0, values + (size_t)row0 * P_DIM);
  }

  for (int k = 0; k < NCHUNK; ++k) {
    if (issuer) {
      if (k + 1 < NCHUNK) {
        // Prefetch next tile into the other buffer, then wait until at most
        // one TDM op (the prefetch) is outstanding => tile k has landed.
        tdm_load_tile((((k + 1) & 1) ? stage_off1 : stage_off0),
                      values + (size_t)row0 * P_DIM + (size_t)(k + 1) * CHUNK);
        __builtin_amdgcn_s_wait_tensorcnt(1);
      } else {
        __builtin_amdgcn_s_wait_tensorcnt(0);
      }
    }
    __syncthreads();  // tile k visible to all 16 waves

    // Thread t owns column k*CHUNK + t: one label fetch, then walk the rows.
    const int    col = k * CHUNK + tid;
    const int    lab = labels[col];
    const float* sp  = &stage[k & 1][tid];
    float*      accp = &acc[lab];

#pragma unroll
    for (int r = 0; r < ROWS; ++r) {
      float v = sp[r * CHUNK];  // bank-conflict-free: lane == bank
      (void)__hip_atomic_fetch_min(accp + r * C_DIM, v,
                                   __ATOMIC_RELAXED,
                                   __HIP_MEMORY_SCOPE_WORKGROUP);
    }
    __syncthreads();  // tile k fully consumed before its buffer is re-filled
  }

  // Vectorized, non-temporal writeback: 2048 float4s by 512 threads.
  // idx4 in [0, 2048): r = idx4 / 128, c = (idx4 % 128) * 4. Both acc reads
  // (ds_load_b128) and out stores (b128, th:NT) are 16B-aligned & coalesced.
  {
    const v4f* accv = (const v4f*)acc;                       // 512 floats/row = 128 v4f/row
    v4f* outv = (v4f*)(out + (size_t)row0 * C_DIM);
#pragma unroll
    for (int it = 0; it < (ROWS * C_DIM / 4) / THREADS; ++it) {  // 4 iters
      const int idx4 = it * THREADS + tid;
      __builtin_nontemporal_store(accv[idx4], &outv[idx4]);
    }
  }
}

extern "C" void kernel_launch(void* const* d_in, const int* in_sizes, int n_in,
                              void* d_out, int out_size, void* d_ws, size_t ws_size,
                              hipStream_t stream) {
  const float* values = (const float*)d_in[0];
  const int*   labels = (const int*)d_in[1];
  float*       out    = (float*)d_out;

  dim3 grid(B_DIM / ROWS);   // 1024 blocks
  dim3 block(THREADS);       // 512 threads = 16 wave32s
  hipLaunchKernelGGL(stratified_min_kernel, grid, block, 0, stream,
                     values, labels, out);
}